// PDFSampler_27925877358608
// MI455X (gfx1250) — compile-verified
//
#include <hip/hip_runtime.h>
#include <stdint.h>

// PDF stratified resampling (NeRF-style), MI455X / gfx1250, wave32.
// One wave per ray; TDM async-stages inputs into LDS; merge replaces sort.

#define RAYS_TOTAL   524288
#define NE           64          // existing intervals / pdf bins
#define NB           65          // cdf entries == existing_bins == samples
#define OUTP         130         // merged output per ray
#define WPB          8           // waves per block
#define HIST_PAD_F   0.01f
#define EPS_F        1e-5f

typedef unsigned int u32x4 __attribute__((ext_vector_type(4)));
typedef int          i32x4 __attribute__((ext_vector_type(4)));
typedef int          i32x8 __attribute__((ext_vector_type(8)));

// Flat LDS pointer -> LDS byte offset (flat addr low 32 bits ARE the LDS addr
// per CDNA5 aperture rules).
__device__ __forceinline__ uint32_t lds_off(const void* p) {
  return (uint32_t)(uintptr_t)p;
}

// Tensor Data Mover: async copy of `n4` 4-byte elements (1-D tile) from
// global memory to LDS. Builds D# group0/group1 per CDNA5 ISA.
__device__ __forceinline__ void tdm_load_1d(uint32_t lds_addr, const void* gp, uint32_t n4) {
  uint64_t ga = (uint64_t)(uintptr_t)gp;
  u32x4 g0;
  g0[0] = 0x1u;                                   // count=1, user descriptor
  g0[1] = lds_addr;                               // lds_addr[31:0]
  g0[2] = (uint32_t)ga;                           // global_addr[31:0]
  g0[3] = ((uint32_t)(ga >> 32) & 0x01FFFFFFu)    // global_addr[56:32]
          | (2u << 30);                           // type = 2 ("image")
  i32x8 g1;
  g1[0] = (int)(2u << 16);                        // data_size=2 -> 4B; mask=0
  g1[1] = (int)((n4 & 0xFFFFu) << 16);            // tensor_dim0[15:0]
  g1[2] = (int)(((n4 >> 16) & 0xFFFFu) | (1u << 16)); // dim0[31:16], dim1=1
  g1[3] = (int)((n4 & 0xFFFFu) << 16);            // dim1 hi=0, tile_dim0=n4
  g1[4] = 0;                                      // tile_dim1=0, tile_dim2=0
  g1[5] = (int)n4;                                // tensor_dim0_stride lo
  g1[6] = 0;
  g1[7] = 0;
  i32x4 gz = {0, 0, 0, 0};
#if defined(__clang_major__) && (__clang_major__ >= 23)
  i32x8 gz8 = {0, 0, 0, 0, 0, 0, 0, 0};
  __builtin_amdgcn_tensor_load_to_lds(g0, g1, gz, gz, gz8, 0);
#else
  __builtin_amdgcn_tensor_load_to_lds(g0, g1, gz, gz, 0);
#endif
}

__global__ __launch_bounds__(256, 2) void pdf_sample_kernel(
    const float* __restrict__ weights,
    const float* __restrict__ starts,
    const float* __restrict__ ends,
    float* __restrict__ out)
{
  __shared__ float lw [WPB][NE];
  __shared__ float ls [WPB][NE];
  __shared__ float le [WPB][NE];
  __shared__ float cdf[WPB][NB + 3];   // +3 pad: bank-conflict stagger
  __shared__ float eb [WPB][NB + 3];
  __shared__ float sb [WPB][NB + 3];

  const int lane = threadIdx.x & 31;
  const int wave = threadIdx.x >> 5;
  const int ray  = blockIdx.x * WPB + wave;

  // ---- Stage this wave's ray inputs via TDM (3 x 256B async DMA -> LDS) ----
  tdm_load_1d(lds_off(&lw[wave][0]), weights + (size_t)ray * NE, NE);
  tdm_load_1d(lds_off(&ls[wave][0]), starts  + (size_t)ray * NE, NE);
  tdm_load_1d(lds_off(&le[wave][0]), ends    + (size_t)ray * NE, NE);
  __builtin_amdgcn_s_wait_tensorcnt(0);
  asm volatile("" ::: "memory");

  // ---- PDF -> clamped CDF (each lane owns bins 2*lane, 2*lane+1) ----
  const int i0 = lane * 2, i1 = i0 + 1;
  float w0 = lw[wave][i0] + HIST_PAD_F;
  float w1 = lw[wave][i1] + HIST_PAD_F;

  float ws = w0 + w1;                       // wave32 tree reduction
  #pragma unroll
  for (int off = 16; off > 0; off >>= 1) ws += __shfl_xor(ws, off, 32);

  float padding = fmaxf(EPS_F - ws, 0.0f);
  w0 += padding * (1.0f / NE);
  w1 += padding * (1.0f / NE);
  ws += padding;
  float p0 = w0 / ws;
  float p1 = w1 / ws;

  float v = p0 + p1;                        // inclusive scan of pair sums
  #pragma unroll
  for (int off = 1; off < 32; off <<= 1) {
    float n = __shfl_up(v, off, 32);
    if (lane >= off) v += n;
  }
  float excl = v - (p0 + p1);
  if (lane == 0) cdf[wave][0] = 0.0f;
  cdf[wave][i0 + 1] = fminf(excl + p0, 1.0f);
  cdf[wave][i1 + 1] = fminf(excl + p0 + p1, 1.0f);

  // ---- Existing bins: [s0, midpoints, e_last] (monotone by construction) ----
  float s0 = ls[wave][i0], s1 = ls[wave][i1];
  float e0 = le[wave][i0], e1 = le[wave][i1];
  if (lane == 0) eb[wave][0] = s0;
  else           eb[wave][i0] = 0.5f * (s0 + le[wave][i0 - 1]);
  eb[wave][i1] = 0.5f * (s1 + e0);
  if (lane == 31) eb[wave][NB - 1] = e1;
  __syncthreads();

  // ---- Inverse-transform sample 65 stratified u's (searchsorted + lerp) ----
  float* cdfp = &cdf[wave][0];
  float* ebp  = &eb[wave][0];
  for (int j = lane; j < NB; j += 32) {
    float u = (float)j * (1.0f / (NB - 1));
    int lo = 0, hi = NB;                    // searchsorted(cdf, u, 'right')
    while (lo < hi) {
      int mid = (lo + hi) >> 1;
      if (cdfp[mid] <= u) lo = mid + 1; else hi = mid;
    }
    int below = min(max(lo - 1, 0), NB - 1);
    int above = min(lo, NB - 1);
    float cg0 = cdfp[below], cg1 = cdfp[above];
    float bg0 = ebp[below],  bg1 = ebp[above];
    float t = (u - cg0) / (cg1 - cg0);
    t = (t != t) ? 0.0f : t;                // nan_to_num(nan=0)
    t = fminf(fmaxf(t, 0.0f), 1.0f);        // clip(0,1) (also tames +/-inf)
    sb[wave][j] = bg0 + t * (bg1 - bg0);
  }
  __syncthreads();

  // ---- "sort(concat)" == merge of two sorted 65-arrays via merge-path ----
  float* sbp  = &sb[wave][0];
  float* orow = out + (size_t)ray * OUTP;
  for (int k = lane; k < OUTP; k += 32) {
    int lo = max(0, k - NB), hi = min(k, NB);
    while (lo < hi) {                       // co-rank of k-th merged element
      int mid = (lo + hi) >> 1;
      if (ebp[mid] <= sbp[k - mid - 1]) lo = mid + 1; else hi = mid;
    }
    int a = lo, b = k - lo;
    float va = (a < NB) ? ebp[a] : 3.4028235e38f;
    float vb = (b < NB) ? sbp[b] : 3.4028235e38f;
    __builtin_nontemporal_store(fminf(va, vb), &orow[k]);  // stream output
  }
}

extern "C" void kernel_launch(void* const* d_in, const int* in_sizes, int n_in,
                              void* d_out, int out_size, void* d_ws, size_t ws_size,
                              hipStream_t stream) {
  const float* weights = (const float*)d_in[0];
  const float* starts  = (const float*)d_in[1];
  const float* ends    = (const float*)d_in[2];
  float* out = (float*)d_out;
  (void)in_sizes; (void)n_in; (void)out_size; (void)d_ws; (void)ws_size;

  dim3 grid(RAYS_TOTAL / WPB);
  dim3 block(256);
  hipLaunchKernelGGL(pdf_sample_kernel, grid, block, 0, stream,
                     weights, starts, ends, out);
}